// AttentionLayer_58420145160277
// MI455X (gfx1250) — compile-verified
//
#include <hip/hip_runtime.h>

typedef __attribute__((ext_vector_type(2))) float v2f;
typedef __attribute__((ext_vector_type(8))) float v8f;

#define DD 256            // D == U == T_en == T_de
#define BB 8
#define TILE_T 8          // decoder rows per block in fused kernel
#define U_CHUNK 32        // u-chunk staged in LDS
#define EPAD 257          // 257 % 64 == 1 -> conflict-free column reads

__device__ __forceinline__ float fast_tanh(float x) {
#if __has_builtin(__builtin_amdgcn_tanhf)
    return __builtin_amdgcn_tanhf(x);          // v_tanh_f32 (gfx1250 HW tanh,
                                               // probe-confirmed on device pass)
#else
    // host-pass / portability fallback: tanh(x) = 1 - 2/(exp2(2x*log2 e)+1)
    float e = __builtin_amdgcn_exp2f(x * 2.8853900817779268f);
    return 1.0f - 2.0f / (e + 1.0f);
#endif
}

// ---------------------------------------------------------------------------
// Kernel A: C[M x 256] = A[M x 256] * W[256 x 256], fp32.
// Each wave computes a 32x64 strip = 2 M-tiles x 4 N-tiles of 16x16, so each
// k4-step feeds 8 v_wmma_f32_16x16x4_f32 from 16B(A)+32B(B) per lane
// (6 B/WMMA instead of 16 B/WMMA for the 1-tile version).
//
// Fragment layouts (wave32):
//  A (16x4): lane<16 -> M=lane,    K={k,k+1};  lane>=16 -> M=lane-16, K={k+2,k+3}
//  B (4x16): n = lane&15; VGPR0 = W[k+2*half][n], VGPR1 = W[k+2*half+1][n]
//  D (16x16): VGPR r -> row = r + 8*half, col = lane&15
// ---------------------------------------------------------------------------
#define GEMM_MT 2         // 16-row tiles per wave  (32 rows)
#define GEMM_NT 4         // 16-col tiles per wave  (64 cols)

__global__ void __launch_bounds__(256)
att_gemm_wmma(const float* __restrict__ A, const float* __restrict__ W,
              float* __restrict__ C) {
    const int wid   = (blockIdx.x * blockDim.x + threadIdx.x) >> 5;
    const int lane  = threadIdx.x & 31;
    const int nTn   = DD / (16 * GEMM_NT);       // 4 strip-cols
    const int sm    = wid / nTn;                 // strip row   (0..63)
    const int sn    = wid % nTn;                 // strip col   (0..3)
    const int half  = lane >> 4;
    const int mr    = lane & 15;
    const int nl    = lane & 15;
    const int row0  = sm * (16 * GEMM_MT);
    const int col0  = sn * (16 * GEMM_NT);

    v8f acc[GEMM_MT][GEMM_NT];
    #pragma unroll
    for (int i = 0; i < GEMM_MT; ++i)
        #pragma unroll
        for (int j = 0; j < GEMM_NT; ++j) acc[i][j] = (v8f){};

    #pragma unroll 2
    for (int k = 0; k < DD; k += 4) {
        const int kk = k + 2 * half;
        v2f a[GEMM_MT], b[GEMM_NT];
        #pragma unroll
        for (int i = 0; i < GEMM_MT; ++i) {
            const float* ap = A + (size_t)(row0 + i * 16 + mr) * DD + kk;
            a[i].x = ap[0];
            a[i].y = ap[1];
        }
        #pragma unroll
        for (int j = 0; j < GEMM_NT; ++j) {
            const float* wp = W + (size_t)kk * DD + col0 + j * 16 + nl;
            b[j].x = wp[0];
            b[j].y = wp[DD];
        }
#if __has_builtin(__builtin_amdgcn_wmma_f32_16x16x4_f32)
        // device pass: v_wmma_f32_16x16x4_f32 (histogram-confirmed)
        #pragma unroll
        for (int i = 0; i < GEMM_MT; ++i)
            #pragma unroll
            for (int j = 0; j < GEMM_NT; ++j)
                acc[i][j] = __builtin_amdgcn_wmma_f32_16x16x4_f32(
                    false, a[i], false, b[j], (short)0, acc[i][j], false, false);
#else
        // host-pass placeholder only (never executed; device pass uses WMMA)
        #pragma unroll
        for (int i = 0; i < GEMM_MT; ++i)
            #pragma unroll
            for (int j = 0; j < GEMM_NT; ++j)
                acc[i][j][0] += a[i].x * b[j].x + a[i].y * b[j].y;
#endif
    }

    #pragma unroll
    for (int i = 0; i < GEMM_MT; ++i)
        #pragma unroll
        for (int j = 0; j < GEMM_NT; ++j)
            #pragma unroll
            for (int r = 0; r < 8; ++r)
                C[(size_t)(row0 + i * 16 + r + 8 * half) * DD +
                  col0 + j * 16 + nl] = acc[i][j][r];
}

// ---------------------------------------------------------------------------
// Kernel B: fused  tanh-score + softmax + weighted-sum + residual.
// grid = B * (T_de/TILE_T) = 256 blocks, 256 threads (8 wave32).
// Block owns (b, t0..t0+7). Thread owns encoder column e = tid.
// ---------------------------------------------------------------------------
__global__ void __launch_bounds__(256)
att_fused(const float* __restrict__ attEnAll,  // (B,256,256) ws
          const float* __restrict__ attDeAll,  // (B,256,256) ws
          const float* __restrict__ enAll,     // en_seq (B,256,256)
          const float* __restrict__ deAll,     // de_seq (B,256,256)
          const float* __restrict__ nu,        // (256)
          float* __restrict__ out) {           // (B,256,256)
    __shared__ float s_en_t[U_CHUNK * EPAD];     // transposed [u][e], padded
    __shared__ float s_de[TILE_T * DD];          // att_de rows (broadcast reads)
    __shared__ float s_nu[DD];
    __shared__ float s_alp[TILE_T * DD];         // mu -> exp values
    __shared__ float s_inv[TILE_T];

    const int tid = threadIdx.x;                 // 0..255
    const int b   = blockIdx.x >> 5;             // 32 t-blocks per batch
    const int t0  = (blockIdx.x & 31) * TILE_T;

    const float* attEn = attEnAll + b * (DD * DD);
    const float* attDe = attDeAll + (b * DD + t0) * DD;

    for (int i = tid; i < TILE_T * DD; i += 256) s_de[i] = attDe[i];
    s_nu[tid] = nu[tid];

    float acc[TILE_T];
    #pragma unroll
    for (int t = 0; t < TILE_T; ++t) acc[t] = 0.0f;

    // ---- mu[t][e] = sum_u tanh(att_en[e,u] + att_de[t,u]) * nu[u] ----
    for (int u0 = 0; u0 < DD; u0 += U_CHUNK) {
        __syncthreads();
        // stage att_en[:, u0:u0+32] transposed; global reads coalesced,
        // LDS writes stride EPAD ≡ 1 mod 64 -> conflict-free
        for (int idx = tid; idx < DD * U_CHUNK; idx += 256) {
            const int u = idx & (U_CHUNK - 1);
            const int e = idx >> 5;
            s_en_t[u * EPAD + e] = attEn[e * DD + u0 + u];
        }
        __syncthreads();
        const int e = tid;
        #pragma unroll 4
        for (int u = 0; u < U_CHUNK; ++u) {
            const float aen = s_en_t[u * EPAD + e];    // conflict-free
            const float nuu = s_nu[u0 + u];            // broadcast
            #pragma unroll
            for (int t = 0; t < TILE_T; ++t) {
                const float v = fast_tanh(aen + s_de[t * DD + u0 + u]);
                acc[t] = fmaf(v, nuu, acc[t]);
            }
        }
    }
    __syncthreads();
    #pragma unroll
    for (int t = 0; t < TILE_T; ++t) s_alp[t * DD + tid] = acc[t];
    __syncthreads();

    // ---- softmax over e: wave w reduces row t = w (wave32 shuffles) ----
    {
        const int w    = tid >> 5;
        const int lane = tid & 31;
        float v[8];
        float vmax = -__FLT_MAX__;
        #pragma unroll
        for (int i = 0; i < 8; ++i) {
            v[i] = s_alp[w * DD + lane + 32 * i];
            vmax = fmaxf(vmax, v[i]);
        }
        #pragma unroll
        for (int off = 16; off > 0; off >>= 1)
            vmax = fmaxf(vmax, __shfl_xor(vmax, off, 32));
        float s = 0.0f;
        #pragma unroll
        for (int i = 0; i < 8; ++i) {
            const float ex =
                __builtin_amdgcn_exp2f((v[i] - vmax) * 1.4426950408889634f);
            s_alp[w * DD + lane + 32 * i] = ex;
            s += ex;
        }
        #pragma unroll
        for (int off = 16; off > 0; off >>= 1)
            s += __shfl_xor(s, off, 32);
        if (lane == 0) s_inv[w] = 1.0f / s;
    }
    __syncthreads();

    // ---- out[t,d] = de[t,d] + (sum_e exp[t,e]*en[e,d]) * inv_sum[t] ----
    {
        const int d = tid;
        const float* enB = enAll + b * (DD * DD);
        float o[TILE_T];
        #pragma unroll
        for (int t = 0; t < TILE_T; ++t) o[t] = 0.0f;
        for (int e = 0; e < DD; ++e) {
            const float x = enB[e * DD + d];           // coalesced, reused 8x
            #pragma unroll
            for (int t = 0; t < TILE_T; ++t)
                o[t] = fmaf(s_alp[t * DD + e], x, o[t]);   // broadcast
        }
        const float* deB = deAll + (b * DD + t0) * DD;
        float* outB      = out + (b * DD + t0) * DD;
        #pragma unroll
        for (int t = 0; t < TILE_T; ++t)
            outB[t * DD + d] = deB[t * DD + d] + o[t] * s_inv[t];
    }
}

extern "C" void kernel_launch(void* const* d_in, const int* in_sizes, int n_in,
                              void* d_out, int out_size, void* d_ws, size_t ws_size,
                              hipStream_t stream) {
    const float* en   = (const float*)d_in[0];  // (8,256,256)
    const float* de   = (const float*)d_in[1];  // (8,256,256)
    const float* w_en = (const float*)d_in[2];  // (256,256)
    const float* w_de = (const float*)d_in[3];  // (256,256)
    const float* nu   = (const float*)d_in[4];  // (256,1)
    float* out = (float*)d_out;

    const int M = BB * DD;                       // 2048 rows
    float* att_en = (float*)d_ws;                // 2 MB
    float* att_de = att_en + (size_t)M * DD;     // 2 MB

    // waves = (M/32) strips * (256/64) strip-cols = 256 waves -> 32 blocks
    const int gemmWaves  = (M / (16 * GEMM_MT)) * (DD / (16 * GEMM_NT));
    const int gemmBlocks = gemmWaves * 32 / 256;
    att_gemm_wmma<<<gemmBlocks, 256, 0, stream>>>(en, w_en, att_en);
    att_gemm_wmma<<<gemmBlocks, 256, 0, stream>>>(de, w_de, att_de);

    // fused scores+softmax+context: 8 batches * 32 t-blocks
    att_fused<<<BB * (DD / TILE_T), 256, 0, stream>>>(att_en, att_de, en, de,
                                                      nu, out);
}